// AttnDecoderRNN_23398981828723
// MI455X (gfx1250) — compile-verified
//
#include <hip/hip_runtime.h>
#include <math.h>

// Problem dims
#define H   256
#define V   50257
#define L   2048
#define B   32
#define K2  512                   // 2*H
#define G   1024                  // 4*H
#define NVT ((V + 15) / 16)       // vocab tiles of 16 -> 3142
#define NGT (G / 16)              // gate-row tiles   -> 64
#define NC  8                     // L-chunks for attn-apply partials
#define LC  (L / NC)              // 256 l per chunk

typedef __attribute__((ext_vector_type(2))) float v2f;
typedef __attribute__((ext_vector_type(8))) float v8f;

#define WMMA_F32(a, b, c) \
    __builtin_amdgcn_wmma_f32_16x16x4_f32(false, (a), false, (b), (short)0, (c), false, false)

// ---------------------------------------------------------------------------
// Kernel 1: u[h] = sum_k We[k,h] * v[k],  We[k,h] = W_attn[k, H+h]
// (softmax shift-invariance removes eh and b_attn from the attention weights)
// ---------------------------------------------------------------------------
__global__ __launch_bounds__(H) void k_attn_vec(const float* __restrict__ W_attn,
                                                const float* __restrict__ vvec,
                                                float* __restrict__ u) {
    int h = threadIdx.x;
    float acc = 0.f;
    for (int k = 0; k < H; ++k)
        acc += W_attn[k * K2 + H + h] * vvec[k];   // coalesced across h
    u[h] = acc;
}

// ---------------------------------------------------------------------------
// Kernel 2: scores[b,l] = dot(enc[b,l,:], u)   (one wave32 per (b,l) row)
// 64 MB HBM stream, ~2.7us roofline; 65536 waves of parallelism.
// ---------------------------------------------------------------------------
__global__ __launch_bounds__(256) void k_scores(const float* __restrict__ enc,
                                                const float* __restrict__ u,
                                                float* __restrict__ scores) {
    int row  = (blockIdx.x * blockDim.x + threadIdx.x) >> 5;
    int lane = threadIdx.x & 31;
    if (row >= B * L) return;
    const float* r = enc + (size_t)row * H;
    float acc = 0.f;
#pragma unroll
    for (int j = 0; j < H / 32; ++j)
        acc += r[lane + 32 * j] * u[lane + 32 * j];
#pragma unroll
    for (int off = 16; off > 0; off >>= 1)        // wave32 butterfly
        acc += __shfl_xor(acc, off, 32);
    if (lane == 0) scores[row] = acc;
}

// ---------------------------------------------------------------------------
// Kernel 3: attn_weights[b,:] = softmax(scores[b,:]) over L (block per b)
// ---------------------------------------------------------------------------
__global__ __launch_bounds__(256) void k_softmax(const float* __restrict__ scores,
                                                 float* __restrict__ w_out) {
    __shared__ float red[256];
    int b = blockIdx.x, t = threadIdx.x;
    const float* s = scores + b * L;
    float m = -INFINITY;
    for (int l = t; l < L; l += 256) m = fmaxf(m, s[l]);
    red[t] = m; __syncthreads();
    for (int st = 128; st > 0; st >>= 1) {
        if (t < st) red[t] = fmaxf(red[t], red[t + st]);
        __syncthreads();
    }
    m = red[0]; __syncthreads();
    float sum = 0.f;
    for (int l = t; l < L; l += 256) sum += __expf(s[l] - m);
    red[t] = sum; __syncthreads();
    for (int st = 128; st > 0; st >>= 1) {
        if (t < st) red[t] += red[t + st];
        __syncthreads();
    }
    float inv = 1.f / red[0];
    for (int l = t; l < L; l += 256) w_out[b * L + l] = __expf(s[l] - m) * inv;
}

// ---------------------------------------------------------------------------
// Kernel 4a: partial attn_applied over an L-chunk. Grid = B*NC blocks so the
// (L2-resident) second pass over encoder_outputs has 8x more concurrency.
// partial[(b*NC + c)*H + h] = sum_{l in chunk c} w[b,l]*enc[b,l,h]
// ---------------------------------------------------------------------------
__global__ __launch_bounds__(256) void k_attn_partial(const float* __restrict__ enc,
                                                      const float* __restrict__ w,
                                                      float* __restrict__ partial) {
    __shared__ float wl[LC];
    int b = blockIdx.x / NC, c = blockIdx.x % NC, t = threadIdx.x;
    wl[t] = w[b * L + c * LC + t];
    __syncthreads();
    const float* eb = enc + ((size_t)b * L + c * LC) * H;
    float acc = 0.f;
    for (int l = 0; l < LC; ++l)
        acc += wl[l] * eb[l * H + t];             // coalesced across t
    partial[(size_t)(b * NC + c) * H + t] = acc;
}

// ---------------------------------------------------------------------------
// Kernel 4b: combine partials; build x = [embedded | attn] and y[...,H:] part.
// ---------------------------------------------------------------------------
__global__ __launch_bounds__(256) void k_attn_combine(const float* __restrict__ partial,
                                                      const int*   __restrict__ tokens,
                                                      const float* __restrict__ emb,
                                                      float* __restrict__ ws_x,
                                                      float* __restrict__ ws_y) {
    int b = blockIdx.x, t = threadIdx.x;
    float acc = 0.f;
#pragma unroll
    for (int c = 0; c < NC; ++c)
        acc += partial[(size_t)(b * NC + c) * H + t];
    ws_x[b * K2 + H + t] = acc;
    ws_y[b * K2 + H + t] = acc;
    ws_x[b * K2 + t] = emb[(size_t)tokens[b] * H + t];
}

// ---------------------------------------------------------------------------
// Kernel 5a: gates = x @ W_ih^T + h_prev @ W_hh^T + b_ih + b_hh via FP32 WMMA.
// Same tiling as the logits GEMM (M = 16 weight rows, N = batch halves),
// so W_ih/W_hh are read exactly once (3 MB) instead of 32x per batch.
// ---------------------------------------------------------------------------
__global__ __launch_bounds__(256) void k_gates_wmma(const float* __restrict__ W_ih,
                                                    const float* __restrict__ b_ih,
                                                    const float* __restrict__ W_hh,
                                                    const float* __restrict__ b_hh,
                                                    const float* __restrict__ ws_x,
                                                    const float* __restrict__ hidden,
                                                    float* __restrict__ gates) {
    int wave = blockIdx.x * (blockDim.x >> 5) + (threadIdx.x >> 5);
    int lane = threadIdx.x & 31;
    if (wave >= NGT) return;                    // wave-uniform exit
    int r0   = wave * 16;
    int half = lane >> 4;                       // K pair select
    int idx  = lane & 15;                       // A: M row / B: N col
    int koff = half << 1;

    const float* ap  = W_ih + (size_t)(r0 + idx) * K2 + koff;
    const float* bp0 = ws_x + (size_t)idx        * K2 + koff;   // batch 0..15
    const float* bp1 = ws_x + (size_t)(idx + 16) * K2 + koff;   // batch 16..31
    v8f c0 = {}; v8f c1 = {};
    for (int k0 = 0; k0 < K2; k0 += 4) {
        v2f a  = *(const v2f*)(ap  + k0);
        v2f b0 = *(const v2f*)(bp0 + k0);
        v2f b1 = *(const v2f*)(bp1 + k0);
        c0 = WMMA_F32(a, b0, c0);
        c1 = WMMA_F32(a, b1, c1);
    }
    const float* ah  = W_hh   + (size_t)(r0 + idx) * H + koff;
    const float* hp0 = hidden + (size_t)idx        * H + koff;
    const float* hp1 = hidden + (size_t)(idx + 16) * H + koff;
    for (int k0 = 0; k0 < H; k0 += 4) {
        v2f a  = *(const v2f*)(ah  + k0);
        v2f b0 = *(const v2f*)(hp0 + k0);
        v2f b1 = *(const v2f*)(hp1 + k0);
        c0 = WMMA_F32(a, b0, c0);
        c1 = WMMA_F32(a, b1, c1);
    }
    int mbase = half << 3;                      // D: VGPR r -> M = r + 8*half
#pragma unroll
    for (int r = 0; r < 8; ++r) {
        int row = r0 + mbase + r;
        float bias = b_ih[row] + b_hh[row];
        gates[(size_t)idx        * G + row] = c0[r] + bias;
        gates[(size_t)(idx + 16) * G + row] = c1[r] + bias;
    }
}

// ---------------------------------------------------------------------------
// Kernel 5b: LSTM elementwise. c_new/h_new from gates (i|f|g|o row blocks).
// ---------------------------------------------------------------------------
__global__ __launch_bounds__(H) void k_lstm_elem(const float* __restrict__ gates,
                                                 const float* __restrict__ cell,
                                                 float* __restrict__ h_out,
                                                 float* __restrict__ c_out,
                                                 float* __restrict__ ws_y) {
    int b = blockIdx.x, j = threadIdx.x;
    const float* gb = gates + (size_t)b * G;
    float ig = 1.f / (1.f + __expf(-gb[j]));
    float fg = 1.f / (1.f + __expf(-gb[H + j]));
    float gg = tanhf(gb[2 * H + j]);
    float og = 1.f / (1.f + __expf(-gb[3 * H + j]));
    float cn = fg * cell[b * H + j] + ig * gg;
    float hn = og * tanhf(cn);
    h_out[b * H + j] = hn;
    c_out[b * H + j] = cn;
    ws_y[b * K2 + j] = hn;                      // y = [h_new | attn_applied]
}

// ---------------------------------------------------------------------------
// Kernel 6: logits = y (B x 512) @ W_out^T (V x 512) + b_out via FP32 WMMA.
// Dominant cost: 103 MB W_out stream -> ~4.4us at 23.3 TB/s. One wave per
// 16-wide vocab tile; two batch halves share each A load; EXEC all-1s in loop.
// ---------------------------------------------------------------------------
__global__ __launch_bounds__(256) void k_logits_wmma(const float* __restrict__ W_out,
                                                     const float* __restrict__ b_out,
                                                     const float* __restrict__ y,
                                                     float* __restrict__ logits) {
    int wave = blockIdx.x * (blockDim.x >> 5) + (threadIdx.x >> 5);
    int lane = threadIdx.x & 31;
    if (wave >= NVT) return;                    // wave-uniform exit
    int v0   = wave * 16;
    int half = lane >> 4;
    int idx  = lane & 15;
    int koff = half << 1;

    int arow = v0 + idx; if (arow > V - 1) arow = V - 1;   // clamp ragged tile
    const float* ap  = W_out + (size_t)arow * K2 + koff;
    const float* bp0 = y + (size_t)idx        * K2 + koff;
    const float* bp1 = y + (size_t)(idx + 16) * K2 + koff;

    v8f c0 = {}; v8f c1 = {};
    for (int k0 = 0; k0 < K2; k0 += 4) {
        v2f a  = *(const v2f*)(ap  + k0);
        v2f b0 = *(const v2f*)(bp0 + k0);
        v2f b1 = *(const v2f*)(bp1 + k0);
        c0 = WMMA_F32(a, b0, c0);
        c1 = WMMA_F32(a, b1, c1);
    }
    int mbase = half << 3;
#pragma unroll
    for (int r = 0; r < 8; ++r) {
        int vcol = v0 + mbase + r;
        if (vcol < V) {
            float bias = b_out[vcol];
            logits[(size_t)idx        * V + vcol] = c0[r] + bias;
            logits[(size_t)(idx + 16) * V + vcol] = c1[r] + bias;
        }
    }
}

// ---------------------------------------------------------------------------
// Kernel 7: in-place log_softmax over V per batch row (block per b)
// ---------------------------------------------------------------------------
__global__ __launch_bounds__(1024) void k_logsoftmax(float* __restrict__ logits) {
    __shared__ float red[1024];
    int b = blockIdx.x, t = threadIdx.x;
    float* row = logits + (size_t)b * V;
    float m = -INFINITY;
    for (int i = t; i < V; i += 1024) m = fmaxf(m, row[i]);
    red[t] = m; __syncthreads();
    for (int st = 512; st > 0; st >>= 1) {
        if (t < st) red[t] = fmaxf(red[t], red[t + st]);
        __syncthreads();
    }
    m = red[0]; __syncthreads();
    float s = 0.f;
    for (int i = t; i < V; i += 1024) s += __expf(row[i] - m);
    red[t] = s; __syncthreads();
    for (int st = 512; st > 0; st >>= 1) {
        if (t < st) red[t] += red[t + st];
        __syncthreads();
    }
    float lse = m + __logf(red[0]);
    for (int i = t; i < V; i += 1024) row[i] = row[i] - lse;  // each thread owns i
}

// ---------------------------------------------------------------------------
extern "C" void kernel_launch(void* const* d_in, const int* in_sizes, int n_in,
                              void* d_out, int out_size, void* d_ws, size_t ws_size,
                              hipStream_t stream) {
    const int*   tokens = (const int*)  d_in[0];
    const float* hidden = (const float*)d_in[1];
    const float* cell   = (const float*)d_in[2];
    const float* enc    = (const float*)d_in[3];
    const float* emb    = (const float*)d_in[4];
    const float* W_attn = (const float*)d_in[5];
    // d_in[6] = b_attn: drops out of softmax exactly (shift invariance)
    const float* vvec   = (const float*)d_in[7];
    const float* W_ih   = (const float*)d_in[8];
    const float* b_ih   = (const float*)d_in[9];
    const float* W_hh   = (const float*)d_in[10];
    const float* b_hh   = (const float*)d_in[11];
    const float* W_out  = (const float*)d_in[12];
    const float* b_out  = (const float*)d_in[13];

    float* out       = (float*)d_out;
    float* log_probs = out;                         // B*V
    float* h_out     = out + (size_t)B * V;         // B*H
    float* c_out     = h_out + B * H;               // B*H
    float* attn_w    = c_out + B * H;               // B*L

    float* ws      = (float*)d_ws;
    float* u       = ws;                  // H
    float* scores  = u + H;               // B*L
    float* ws_x    = scores + B * L;      // B*K2  ([embedded | attn])
    float* ws_y    = ws_x + B * K2;       // B*K2  ([h_new    | attn])
    float* gates   = ws_y + B * K2;       // B*G
    float* partial = gates + B * G;       // B*NC*H

    k_attn_vec    <<<1, H, 0, stream>>>(W_attn, vvec, u);
    k_scores      <<<(B * L * 32 + 255) / 256, 256, 0, stream>>>(enc, u, scores);
    k_softmax     <<<B, 256, 0, stream>>>(scores, attn_w);
    k_attn_partial<<<B * NC, 256, 0, stream>>>(enc, attn_w, partial);
    k_attn_combine<<<B, 256, 0, stream>>>(partial, tokens, emb, ws_x, ws_y);
    k_gates_wmma  <<<(NGT + 7) / 8, 256, 0, stream>>>(W_ih, b_ih, W_hh, b_hh,
                                                      ws_x, hidden, gates);
    k_lstm_elem   <<<B, H, 0, stream>>>(gates, cell, h_out, c_out, ws_y);
    k_logits_wmma <<<(NVT + 7) / 8, 256, 0, stream>>>(W_out, b_out, ws_y, log_probs);
    k_logsoftmax  <<<B, 1024, 0, stream>>>(log_probs);
}